// TextEncoder_10642928960013
// MI455X (gfx1250) — compile-verified
//
#include <hip/hip_runtime.h>
#include <math.h>

#define BB 32
#define EPS 1e-5f
#define PROP_F 0.69813170079773183f   // 40/180*pi

typedef float v2f __attribute__((ext_vector_type(2)));
typedef float v8f __attribute__((ext_vector_type(8)));

// ---------------- e[b,s] = sum_h x[b,s,h]^2  (one wave per row) ----------------
__global__ void e_kernel(const float* __restrict__ x, float* __restrict__ e) {
  int wave = (blockIdx.x * blockDim.x + threadIdx.x) >> 5;
  int lane = threadIdx.x & 31;
  if (wave >= BB * 128) return;
  const float* row = x + (size_t)wave * 512;
  float s = 0.f;
  for (int h = lane; h < 512; h += 32) { float v = row[h]; s += v * v; }
  #pragma unroll
  for (int m = 16; m > 0; m >>= 1) s += __shfl_xor(s, m, 32);
  if (lane == 0) e[wave] = s;
}

// ---------------- h0[b,c,y,x] = e[b,c] * p_l[c,y] * p_r[c,x] ----------------
__global__ void h0_kernel(const float* __restrict__ e, const float* __restrict__ pl,
                          const float* __restrict__ pr, float* __restrict__ h0) {
  int idx = blockIdx.x * blockDim.x + threadIdx.x;
  if (idx >= BB * 128 * 32 * 32) return;
  int xx = idx & 31; int t = idx >> 5;
  int yy = t & 31;   t >>= 5;
  int c  = t & 127;  int b = t >> 7;
  h0[idx] = e[b * 128 + c] * pl[c * 32 + yy] * pr[c * 32 + xx];
}

// ---------------- routing: f[b,c] = mean(depthwise3x3(in)) ----------------
__global__ void dwc_mean_kernel(const float* __restrict__ in, const float* __restrict__ dwc_w,
                                float* __restrict__ f, int cin, int H, int W) {
  int c = blockIdx.x, b = blockIdx.y;
  const float* img = in + ((size_t)b * cin + c) * H * W;
  float w[9];
  #pragma unroll
  for (int i = 0; i < 9; ++i) w[i] = dwc_w[c * 9 + i];
  float s = 0.f;
  int npix = H * W;
  for (int p = threadIdx.x; p < npix; p += blockDim.x) {
    int y = p / W, xx = p % W;
    float acc = 0.f;
    #pragma unroll
    for (int tap = 0; tap < 9; ++tap) {
      int yy = y + tap / 3 - 1, xc = xx + tap % 3 - 1;
      if (yy >= 0 && yy < H && xc >= 0 && xc < W) acc += img[yy * W + xc] * w[tap];
    }
    s += acc;
  }
  __shared__ float red[256];
  red[threadIdx.x] = s; __syncthreads();
  for (int st = 128; st > 0; st >>= 1) {
    if (threadIdx.x < st) red[threadIdx.x] += red[threadIdx.x + st];
    __syncthreads();
  }
  if (threadIdx.x == 0) f[b * cin + c] = red[0] / (float)npix;
}

// ------- routing: LN+ReLU, FCs, sigmoid/softsign, rotation matrices * alpha -------
__global__ void routing_kernel(const float* __restrict__ fin,
                               const float* __restrict__ ln_g, const float* __restrict__ ln_b,
                               const float* __restrict__ fa_w, const float* __restrict__ fa_b,
                               const float* __restrict__ ft_w,
                               float* __restrict__ rot, int cin) {
  int b = blockIdx.x, tid = threadIdx.x;   // 128 threads, cin <= 128 (power of two)
  __shared__ float sf[128];
  __shared__ float sred[128];
  __shared__ float sab[8];                 // [0..3]=alpha, [4..7]=theta
  float v = (tid < cin) ? fin[b * cin + tid] : 0.f;
  sred[tid] = v; __syncthreads();
  for (int st = 64; st > 0; st >>= 1) { if (tid < st) sred[tid] += sred[tid + st]; __syncthreads(); }
  float mu = sred[0] / (float)cin;
  __syncthreads();
  float d = (tid < cin) ? (v - mu) : 0.f;
  sred[tid] = d * d; __syncthreads();
  for (int st = 64; st > 0; st >>= 1) { if (tid < st) sred[tid] += sred[tid + st]; __syncthreads(); }
  float var = sred[0] / (float)cin;
  __syncthreads();
  float fl = 0.f;
  if (tid < cin) fl = fmaxf(0.f, d * rsqrtf(var + EPS) * ln_g[tid] + ln_b[tid]);
  sf[tid] = fl; __syncthreads();
  if (tid < 8) {
    int n = tid & 3;
    const float* wv = (tid < 4) ? (fa_w + n * cin) : (ft_w + n * cin);
    float dot = 0.f;
    for (int c2 = 0; c2 < cin; ++c2) dot += sf[c2] * wv[c2];
    if (tid < 4) sab[tid] = 1.f / (1.f + expf(-(dot + fa_b[n])));
    else         sab[tid] = (dot / (1.f + fabsf(dot))) * PROP_F;
  }
  __syncthreads();
  for (int idx = tid; idx < 4 * 81; idx += blockDim.x) {
    int n = idx / 81, rem = idx % 81, o = rem / 9, i = rem % 9;
    float th = sab[4 + n];
    float cs = cosf(th), sn = sinf(th);
    float px  = (float)(o % 3) - 1.f, py  = (float)(o / 3) - 1.f;
    float qx = fminf(1.f, fmaxf(-1.f, px * cs - py * sn));
    float qy = fminf(1.f, fmaxf(-1.f, px * sn + py * cs));
    float pxi = (float)(i % 3) - 1.f, pyi = (float)(i / 3) - 1.f;
    float wx = fmaxf(0.f, 1.f - fabsf(qx - pxi));
    float wy = fmaxf(0.f, 1.f - fabsf(qy - pyi));
    rot[((b * 4 + n) * 9 + o) * 9 + i] = wx * wy * sab[n];
  }
}

// -------- wb[b,co,ci,o] = sum_{n,i} rot[b,n,o,i] * w[n,co,ci,i] --------
__global__ void blend_kernel(const float* __restrict__ rot, const float* __restrict__ w,
                             float* __restrict__ wb, int cout, int cin) {
  int idx = blockIdx.x * blockDim.x + threadIdx.x;
  int total = BB * cout * cin * 9;
  if (idx >= total) return;
  int o = idx % 9; int t = idx / 9;
  int ci = t % cin; t /= cin;
  int co = t % cout; int b = t / cout;
  float s = 0.f;
  #pragma unroll
  for (int n = 0; n < 4; ++n) {
    const float* rr = rot + ((b * 4 + n) * 9 + o) * 9;
    const float* ww = w + (((size_t)n * cout + co) * cin + ci) * 9;
    #pragma unroll
    for (int i = 0; i < 9; ++i) s += rr[i] * ww[i];
  }
  wb[idx] = s;
}

// ---- per-sample 3x3 conv as implicit GEMM on V_WMMA_F32_16X16X4_F32, bn_relu fused ----
// One wave = MBLK x 16(cout) x 16(pixel) tile; K = cin*9 stepped by 4.
// B im2col fetch is amortized over MBLK WMMAs; tap decomposition uses running
// counters (no div/mod in the hot loop).
template <int MBLK>
__global__ void __launch_bounds__(128) arc_conv_wmma(
    const float* __restrict__ in, const float* __restrict__ wb,
    const float* __restrict__ bias,
    const float* __restrict__ bg, const float* __restrict__ bb_,
    const float* __restrict__ bm, const float* __restrict__ bv,
    float* __restrict__ out, int cin, int cout, int H, int W) {
  const int lane = threadIdx.x & 31;
  const int wave = threadIdx.x >> 5;
  const int npix = H * W;
  const int pixTiles = npix >> 4;
  int tile = blockIdx.x * 4 + wave;           // uniform per wave -> EXEC all-ones
  if (tile >= pixTiles) return;
  const int b = blockIdx.z;
  const int co0 = (blockIdx.y * MBLK) << 4;
  const int p0 = tile << 4;
  const int K = cin * 9;
  const int r16 = lane & 15;
  const int khalf = (lane >> 4) << 1;         // lanes 16-31 hold K+2,K+3

  const float* wrow[MBLK];
  #pragma unroll
  for (int m = 0; m < MBLK; ++m)
    wrow[m] = wb + ((size_t)b * cout + (co0 + (m << 4) + r16)) * K + khalf;

  int p = p0 + r16;                           // tiles never cross rows (W % 16 == 0)
  int y = p / W, x = p % W;
  const float* inb = in + (size_t)b * cin * npix;

  // per-lane running decomposition of k = k0 + khalf into (ci, tap)
  int tap = khalf;                            // 0 or 2
  int ci = 0;

  v8f acc[MBLK];
  #pragma unroll
  for (int m = 0; m < MBLK; ++m) acc[m] = (v8f){0.f,0.f,0.f,0.f,0.f,0.f,0.f,0.f};

  #pragma unroll 2
  for (int k0 = 0; k0 < K; k0 += 4) {
    int tap1 = tap + 1, ci1 = ci;
    if (tap1 >= 9) { tap1 -= 9; ci1 += 1; }
    v2f bmat;
    {
      int dy = (tap >= 3) + (tap >= 6);
      int dx = tap - 3 * dy;
      int yy = y + dy - 1, xc = x + dx - 1;
      bmat.x = ((unsigned)yy < (unsigned)H && (unsigned)xc < (unsigned)W)
                   ? inb[(size_t)ci * npix + yy * W + xc] : 0.f;
    }
    {
      int dy = (tap1 >= 3) + (tap1 >= 6);
      int dx = tap1 - 3 * dy;
      int yy = y + dy - 1, xc = x + dx - 1;
      bmat.y = ((unsigned)yy < (unsigned)H && (unsigned)xc < (unsigned)W)
                   ? inb[(size_t)ci1 * npix + yy * W + xc] : 0.f;
    }
    #pragma unroll
    for (int m = 0; m < MBLK; ++m) {
      v2f a;
      a.x = wrow[m][k0];
      a.y = wrow[m][k0 + 1];
      acc[m] = __builtin_amdgcn_wmma_f32_16x16x4_f32(false, a, false, bmat,
                                                     (short)0, acc[m], false, false);
    }
    tap += 4;
    if (tap >= 9) { tap -= 9; ci += 1; }
  }

  // C/D layout: VGPR r -> M = (lane>>4)*8 + r, N = lane&15
  int nIdx = p0 + r16;
  int mbase = (lane >> 4) << 3;
  #pragma unroll
  for (int m = 0; m < MBLK; ++m) {
    #pragma unroll
    for (int r = 0; r < 8; ++r) {
      int co = co0 + (m << 4) + mbase + r;
      float sc = bg[co] * rsqrtf(bv[co] + EPS);
      float val = (acc[m][r] + bias[co]) * sc + (bb_[co] - bm[co] * sc);
      out[((size_t)b * cout + co) * npix + nIdx] = fmaxf(val, 0.f);
    }
  }
}

// ---- conv_transpose 2x2 stride2: out[Y,X] = bias + sum_ci in[Y>>1,X>>1]*w[ci,co,Y&1,X&1], bn_relu ----
__global__ void upconv_bnrelu_kernel(const float* __restrict__ in,
    const float* __restrict__ w, const float* __restrict__ bias,
    const float* __restrict__ bg, const float* __restrict__ bb_,
    const float* __restrict__ bm, const float* __restrict__ bv,
    float* __restrict__ out, int cin, int cout, int Hin, int Win) {
  int Wout = Win << 1, Hout = Hin << 1;
  size_t idx = (size_t)blockIdx.x * blockDim.x + threadIdx.x;
  size_t total = (size_t)BB * cout * Hout * Wout;
  if (idx >= total) return;
  int X = (int)(idx % Wout); size_t t = idx / Wout;
  int Y = (int)(t % Hout);   t /= Hout;
  int co = (int)(t % cout);  int b = (int)(t / cout);
  int y = Y >> 1, x = X >> 1, ky = Y & 1, kx = X & 1;
  const float* ip = in + ((size_t)b * cin) * Hin * Win + (size_t)y * Win + x;
  const float* wp = w + co * 4 + ky * 2 + kx;      // layout [ci][cout][2][2]
  float s = bias[co];
  for (int ci = 0; ci < cin; ++ci)
    s += ip[(size_t)ci * Hin * Win] * wp[(size_t)ci * cout * 4];
  float sc = bg[co] * rsqrtf(bv[co] + EPS);
  float val = s * sc + (bb_[co] - bm[co] * sc);
  out[idx] = fmaxf(val, 0.f);
}

// ---- fused: up2(32->16, 128->256) + bn8_relu + 1x1 head(16->1) + bn9_relu -> d_out ----
__global__ void up2_head_kernel(const float* __restrict__ in, // [32,32,128,128]
    const float* __restrict__ upw, const float* __restrict__ upb,
    const float* __restrict__ g8, const float* __restrict__ b8,
    const float* __restrict__ m8, const float* __restrict__ v8_,
    const float* __restrict__ headw, const float* __restrict__ headb,
    const float* __restrict__ g9, const float* __restrict__ b9,
    const float* __restrict__ m9, const float* __restrict__ v9,
    float* __restrict__ out) {
  int idx = blockIdx.x * blockDim.x + threadIdx.x;
  if (idx >= BB * 256 * 256) return;
  int X = idx & 255; int t = idx >> 8;
  int Y = t & 255;   int b = t >> 8;
  int y = Y >> 1, x = X >> 1, ky = Y & 1, kx = X & 1;
  const float* ip = in + ((size_t)b * 32) * 128 * 128 + (size_t)y * 128 + x;
  float acc[16];
  #pragma unroll
  for (int c = 0; c < 16; ++c) acc[c] = upb[c];
  for (int ci = 0; ci < 32; ++ci) {
    float v = ip[(size_t)ci * 128 * 128];
    const float* wp = upw + ci * 64 + ky * 2 + kx;  // [ci][16][2][2]
    #pragma unroll
    for (int c = 0; c < 16; ++c) acc[c] += v * wp[c * 4];
  }
  float hs = headb[0];
  #pragma unroll
  for (int c = 0; c < 16; ++c) {
    float sc = g8[c] * rsqrtf(v8_[c] + EPS);
    float u = fmaxf(0.f, acc[c] * sc + (b8[c] - m8[c] * sc));
    hs += headw[c] * u;
  }
  float sc9 = g9[0] * rsqrtf(v9[0] + EPS);
  out[idx] = fmaxf(0.f, hs * sc9 + (b9[0] - m9[0] * sc9));
}

extern "C" void kernel_launch(void* const* d_in, const int* in_sizes, int n_in,
                              void* d_out, int out_size, void* d_ws, size_t ws_size,
                              hipStream_t stream) {
  (void)in_sizes; (void)n_in; (void)out_size; (void)ws_size;
  // depth-first insertion-order flattening of setup_inputs()
  const float* x   = (const float*)d_in[0];
  const float* p_l = (const float*)d_in[1];
  const float* p_r = (const float*)d_in[2];
  auto arcp = [&](int L, int j) { return (const float*)d_in[3 + L * 8 + j]; };
  // arc j: 0 weight, 1 bias, 2 dwc_w, 3 ln_g, 4 ln_b, 5 fc_a_w, 6 fc_a_b, 7 fc_t_w
  auto bnp = [&](int i, int j) { return (const float*)d_in[51 + i * 4 + j]; };  // g,b,m,v
  auto upp = [&](int i, int j) { return (const float*)d_in[91 + i * 2 + j]; };  // w,b
  const float* head_w = (const float*)d_in[97];
  const float* head_b = (const float*)d_in[98];

  float* ws = (float*)d_ws;
  const size_t ACT = 16u * 1024u * 1024u;   // 16,777,216 floats = max activation [32,32,128,128]
  float* bufA   = ws;
  float* bufB   = ws + ACT;
  float* wbuf   = ws + 2 * ACT;             // blended weights, max 2,359,296 floats
  float* fbuf   = wbuf + 2400000;           // routing features, max 4096
  float* rotbuf = fbuf + 4096;              // 32*4*81 = 10368
  float* ebuf   = rotbuf + 10400;           // 4096
  // total ~35.97M floats ~= 144 MB of d_ws

  e_kernel<<<(BB * 128 * 32 + 255) / 256, 256, 0, stream>>>(x, ebuf);
  h0_kernel<<<(BB * 128 * 32 * 32 + 255) / 256, 256, 0, stream>>>(ebuf, p_l, p_r, bufA);

  auto run_arc = [&](int L, int cin, int cout, int H, int bn,
                     const float* inp, float* outp) {
    int W = H;
    dwc_mean_kernel<<<dim3(cin, BB), 256, 0, stream>>>(inp, arcp(L, 2), fbuf, cin, H, W);
    routing_kernel<<<BB, 128, 0, stream>>>(fbuf, arcp(L, 3), arcp(L, 4),
                                           arcp(L, 5), arcp(L, 6), arcp(L, 7), rotbuf, cin);
    int total = BB * cout * cin * 9;
    blend_kernel<<<(total + 255) / 256, 256, 0, stream>>>(rotbuf, arcp(L, 0), wbuf, cout, cin);
    int pixTiles = (H * W) >> 4;
    int mtiles = cout >> 4;
    int mblk = (mtiles >= 4) ? 4 : mtiles;   // 1, 2 or 4
    dim3 grid((pixTiles + 3) / 4, mtiles / mblk, BB);
    if (mblk == 4) {
      arc_conv_wmma<4><<<grid, 128, 0, stream>>>(inp, wbuf, arcp(L, 1),
          bnp(bn, 0), bnp(bn, 1), bnp(bn, 2), bnp(bn, 3), outp, cin, cout, H, W);
    } else if (mblk == 2) {
      arc_conv_wmma<2><<<grid, 128, 0, stream>>>(inp, wbuf, arcp(L, 1),
          bnp(bn, 0), bnp(bn, 1), bnp(bn, 2), bnp(bn, 3), outp, cin, cout, H, W);
    } else {
      arc_conv_wmma<1><<<grid, 128, 0, stream>>>(inp, wbuf, arcp(L, 1),
          bnp(bn, 0), bnp(bn, 1), bnp(bn, 2), bnp(bn, 3), outp, cin, cout, H, W);
    }
  };
  auto run_up = [&](int U, int cin, int cout, int Hin, int bn,
                    const float* inp, float* outp) {
    size_t total = (size_t)BB * cout * (Hin * 2) * (Hin * 2);
    upconv_bnrelu_kernel<<<(unsigned)((total + 255) / 256), 256, 0, stream>>>(
        inp, upp(U, 0), upp(U, 1),
        bnp(bn, 0), bnp(bn, 1), bnp(bn, 2), bnp(bn, 3), outp, cin, cout, Hin, Hin);
  };

  run_arc(0, 128,  64,  32, 0, bufA, bufB);
  run_arc(1,  64, 128,  32, 1, bufB, bufA);
  run_up (0, 128,  64,  32, 2, bufA, bufB);
  run_arc(2,  64,  32,  64, 3, bufB, bufA);
  run_arc(3,  32,  64,  64, 4, bufA, bufB);
  run_up (1,  64,  32,  64, 5, bufB, bufA);
  run_arc(4,  32,  16, 128, 6, bufA, bufB);
  run_arc(5,  16,  32, 128, 7, bufB, bufA);
  up2_head_kernel<<<(BB * 256 * 256 + 255) / 256, 256, 0, stream>>>(
      bufA, upp(2, 0), upp(2, 1),
      bnp(8, 0), bnp(8, 1), bnp(8, 2), bnp(8, 3), head_w, head_b,
      bnp(9, 0), bnp(9, 1), bnp(9, 2), bnp(9, 3), (float*)d_out);
}